// GAT_22849226015375
// MI455X (gfx1250) — compile-verified
//
#include <hip/hip_runtime.h>
#include <hip/hip_bf16.h>
#include <math.h>

typedef __attribute__((ext_vector_type(2))) float v2f;
typedef __attribute__((ext_vector_type(8))) float v8f;

#define NEGV (-1000000000000.0f)

__device__ inline v8f wmma_f32_16x16x4(v2f a, v2f b, v8f c) {
  // D = A(16x4, f32) x B(4x16, f32) + C(16x16, f32)
  return __builtin_amdgcn_wmma_f32_16x16x4_f32(
      /*neg_a=*/false, a, /*neg_b=*/false, b,
      /*c_mod=*/(short)0, c, /*reuse_a=*/false, /*reuse_b=*/false);
}

// ---------------------------------------------------------------------------
// Masked instance norm: per (batch, feature) reduce over n nodes.
// grid.x = bs * F, block = 256
// ---------------------------------------------------------------------------
__global__ __launch_bounds__(256) void masked_norm(
    const float* __restrict__ x, const float* __restrict__ mask,
    float* __restrict__ out, int n, int F) {
  int b = blockIdx.x / F;
  int f = blockIdx.x % F;
  const float* xb = x + ((size_t)b * n) * F + f;
  const float* mb = mask + (size_t)b * n;
  __shared__ float red[256];
  __shared__ float redc[256];
  int tid = threadIdx.x;

  float s = 0.f, c = 0.f;
  for (int i = tid; i < n; i += 256) {
    float m = mb[i];
    s += m * xb[(size_t)i * F];
    c += m;
  }
  red[tid] = s; redc[tid] = c;
  __syncthreads();
  for (int st = 128; st > 0; st >>= 1) {
    if (tid < st) { red[tid] += red[tid + st]; redc[tid] += redc[tid + st]; }
    __syncthreads();
  }
  float cnt = redc[0];
  float denom = cnt > 0.f ? cnt : 1.f;
  float mean = red[0] / denom;
  __syncthreads();

  float v = 0.f;
  for (int i = tid; i < n; i += 256) {
    float m = mb[i];
    float d = xb[(size_t)i * F] - mean;
    v += m * d * d;
  }
  red[tid] = v;
  __syncthreads();
  for (int st = 128; st > 0; st >>= 1) {
    if (tid < st) red[tid] += red[tid + st];
    __syncthreads();
  }
  float var = red[0] / denom;
  float inv = rsqrtf(var + 1e-5f);

  float* ob = out + ((size_t)b * n) * F + f;
  for (int i = tid; i < n; i += 256) {
    ob[(size_t)i * F] = (xb[(size_t)i * F] - mean) * inv;
  }
}

// ---------------------------------------------------------------------------
// Generic batched GEMM, N fixed = 64, fp32 via V_WMMA_F32_16X16X4_F32.
// C[y] (MxN) = A[b] (MxK) @ B[b,h] (KxN) (+ bias), y = b*H + h.
// B += b*bStrideB + h*bStrideH (covers shared weights and per-batch B).
// grid = (M/128, bs*H), block = 256 (8 waves, 16 rows/wave).
// ---------------------------------------------------------------------------
__global__ __launch_bounds__(256) void wmma_gemm_n64(
    const float* __restrict__ Abase, const float* __restrict__ Bbase,
    float* __restrict__ Cbase, const float* __restrict__ bias,
    int M, int K, int H, size_t bStrideB, size_t bStrideH) {
  int y = blockIdx.y;
  int b = y / H, h = y % H;
  const float* A = Abase + (size_t)b * M * K;
  const float* B = Bbase + (size_t)b * bStrideB + (size_t)h * bStrideH;
  float* C = Cbase + (size_t)y * M * 64;

  int lane = threadIdx.x & 31;
  int wave = threadIdx.x >> 5;
  int lo = lane & 15, hi = lane >> 4;
  int rowBase = blockIdx.x * 128 + wave * 16;

  v8f acc[4] = {};
  const float* arow = A + (size_t)(rowBase + lo) * K;
  for (int kk = 0; kk < K; kk += 4) {
    // A-frag (ISA 16x4 f32 layout): lanes 0-15 K=kk,kk+1; lanes 16-31 K=kk+2,kk+3
    v2f a;
    a.x = arow[kk + 2 * hi];
    a.y = arow[kk + 2 * hi + 1];
    const float* b0 = B + (size_t)(kk + 2 * hi) * 64 + lo;
#pragma unroll
    for (int nt = 0; nt < 4; ++nt) {
      v2f bb;
      bb.x = b0[nt * 16];
      bb.y = b0[64 + nt * 16];
      acc[nt] = wmma_f32_16x16x4(a, bb, acc[nt]);
    }
  }
#pragma unroll
  for (int nt = 0; nt < 4; ++nt) {
#pragma unroll
    for (int r = 0; r < 8; ++r) {
      int row = rowBase + r + 8 * hi;   // C/D frag: VGPR r -> M = r + 8*hi
      int col = nt * 16 + lo;
      float v = acc[nt][r];
      if (bias) v += bias[col];
      C[(size_t)row * 64 + col] = v;
    }
  }
}

// ---------------------------------------------------------------------------
// s[r] = hp[r,:] . a_src[h],  d[r] = hp[r,:] . a_dst[h]
// ---------------------------------------------------------------------------
__global__ __launch_bounds__(256) void attn_sd(
    const float* __restrict__ hp, const float* __restrict__ asrc,
    const float* __restrict__ adst, float* __restrict__ s,
    float* __restrict__ d, int rows, int n, int H) {
  int r = blockIdx.x * 256 + threadIdx.x;
  if (r >= rows) return;
  int h = (r / n) % H;
  const float* row = hp + (size_t)r * 64;
  const float* as = asrc + h * 64;
  const float* ad = adst + h * 64;
  float ss = 0.f, dd = 0.f;
#pragma unroll
  for (int k = 0; k < 64; ++k) { float v = row[k]; ss += v * as[k]; dd += v * ad[k]; }
  s[r] = ss;
  d[r] = dd;
}

// ---------------------------------------------------------------------------
// Layer-1 fused flash softmax + (attn @ hp) + bias + transpose + ELU.
// One wave per 16-row i-tile; streams j in 64-wide tiles; P staged in LDS
// then consumed as WMMA A-frags against hp B-frags from global (L2-resident).
// grid = bs*H*(n/16), block = 32 (one wave).
// ---------------------------------------------------------------------------
__global__ __launch_bounds__(32) void gat1_flash(
    const float* __restrict__ hp,   // (bs,H,n,64)
    const float* __restrict__ sA,   // (bs,H,n)
    const float* __restrict__ dA,   // (bs,H,n)
    const int* __restrict__ adj,    // (bs,1,n,n)
    const float* __restrict__ bias, // (64)
    float* __restrict__ h2in,       // (bs,n,H*64)  ELU(out) in transposed layout
    int n, int H) {
  __shared__ float P[16][65];       // 65 pad -> conflict-free row-major access
  __shared__ float scl[16];
  __shared__ float ssum[16];

  int lane = threadIdx.x;
  int lo = lane & 15, hi = lane >> 4;
  int tilesPerBH = n / 16;
  int it = blockIdx.x % tilesPerBH;
  int bh = blockIdx.x / tilesPerBH;
  int h = bh % H, b = bh / H;
  int i0 = it * 16;

  const float* srow = sA + (size_t)bh * n;
  const float* drow = dA + (size_t)bh * n;
  const float* hpB = hp + (size_t)bh * n * 64;
  const int* adjRow = adj + ((size_t)b * n + (i0 + lo)) * n;
  float si = srow[i0 + lo];

  v8f acc[4] = {};
  float Mr = -INFINITY, Sr = 0.f;

  for (int jb = 0; jb < n; jb += 64) {
    int j0 = jb + hi * 32;          // lane covers row (i0+lo), cols j0..j0+31
    float p[32];
    float mloc = -INFINITY;
#pragma unroll
    for (int t = 0; t < 32; ++t) {
      float l = si + drow[j0 + t];
      l = l >= 0.f ? l : 0.2f * l;                 // leaky ReLU
      l = adjRow[j0 + t] ? l : NEGV;               // adjacency mask
      p[t] = l;
      mloc = fmaxf(mloc, l);
    }
    mloc = fmaxf(mloc, __shfl_xor(mloc, 16));      // combine half-wave stats
    float mnew = fmaxf(Mr, mloc);
    float scale = __expf(Mr - mnew);               // 0 on first tile (Mr=-inf)
    float psum = 0.f;
#pragma unroll
    for (int t = 0; t < 32; ++t) { p[t] = __expf(p[t] - mnew); psum += p[t]; }
    psum += __shfl_xor(psum, 16);
    Sr = Sr * scale + psum;
    Mr = mnew;

#pragma unroll
    for (int t = 0; t < 32; ++t) P[lo][hi * 32 + t] = p[t];
    if (hi == 0) scl[lo] = scale;
    __syncthreads();

    // rescale accumulator rows by exp(m_old - m_new)
    float rs[8];
#pragma unroll
    for (int r = 0; r < 8; ++r) rs[r] = scl[r + 8 * hi];
#pragma unroll
    for (int nt = 0; nt < 4; ++nt)
#pragma unroll
      for (int r = 0; r < 8; ++r) acc[nt][r] *= rs[r];

    // acc += P(16x64) @ hp[jb:jb+64, 0:64] via 16x16x4 f32 WMMA
    const float* hpj = hpB + (size_t)jb * 64;
#pragma unroll
    for (int kk = 0; kk < 64; kk += 4) {
      v2f a;
      a.x = P[lo][kk + 2 * hi];
      a.y = P[lo][kk + 2 * hi + 1];
      const float* bp = hpj + (size_t)(kk + 2 * hi) * 64 + lo;
#pragma unroll
      for (int nt = 0; nt < 4; ++nt) {
        v2f bb;
        bb.x = bp[nt * 16];
        bb.y = bp[64 + nt * 16];
        acc[nt] = wmma_f32_16x16x4(a, bb, acc[nt]);
      }
    }
    __syncthreads();
  }

  if (hi == 0) ssum[lo] = Sr;
  __syncthreads();

  int F2 = 64 * H;
#pragma unroll
  for (int nt = 0; nt < 4; ++nt) {
#pragma unroll
    for (int r = 0; r < 8; ++r) {
      int row = i0 + r + 8 * hi;
      int col = nt * 16 + lo;
      float v = acc[nt][r] / ssum[r + 8 * hi] + bias[col];
      v = v > 0.f ? v : __expf(v) - 1.f;           // ELU
      h2in[((size_t)b * n + row) * F2 + h * 64 + col] = v;
    }
  }
}

// ---------------------------------------------------------------------------
// Layer-2 row softmax stats: per (b,i) row-max and row-sumexp over j.
// grid = bs*n, block = 256
// ---------------------------------------------------------------------------
__global__ __launch_bounds__(256) void attn2_stats(
    const float* __restrict__ s2, const float* __restrict__ d2,
    const int* __restrict__ adj, float* __restrict__ Mout,
    float* __restrict__ Sout, int n) {
  int b = blockIdx.x / n;
  int i = blockIdx.x % n;
  float si = s2[(size_t)b * n + i];
  const float* db = d2 + (size_t)b * n;
  const int* arow = adj + ((size_t)b * n + i) * n;
  __shared__ float red[256];
  int tid = threadIdx.x;

  float m = -INFINITY;
  for (int j = tid; j < n; j += 256) {
    float l = si + db[j];
    l = l >= 0.f ? l : 0.2f * l;
    l = arow[j] ? l : NEGV;
    m = fmaxf(m, l);
  }
  red[tid] = m;
  __syncthreads();
  for (int st = 128; st > 0; st >>= 1) {
    if (tid < st) red[tid] = fmaxf(red[tid], red[tid + st]);
    __syncthreads();
  }
  m = red[0];
  __syncthreads();

  float ssum = 0.f;
  for (int j = tid; j < n; j += 256) {
    float l = si + db[j];
    l = l >= 0.f ? l : 0.2f * l;
    l = arow[j] ? l : NEGV;
    ssum += __expf(l - m);
  }
  red[tid] = ssum;
  __syncthreads();
  for (int st = 128; st > 0; st >>= 1) {
    if (tid < st) red[tid] += red[tid + st];
    __syncthreads();
  }
  if (tid == 0) {
    Mout[(size_t)b * n + i] = m;
    Sout[(size_t)b * n + i] = red[0];
  }
}

// ---------------------------------------------------------------------------
// Materialize layer-2 attention into d_out (it is a required output).
// ---------------------------------------------------------------------------
__global__ __launch_bounds__(256) void attn2_write(
    const float* __restrict__ s2, const float* __restrict__ d2,
    const int* __restrict__ adj, const float* __restrict__ Mrow,
    const float* __restrict__ Srow, float* __restrict__ attnOut, int n) {
  size_t idx = (size_t)blockIdx.x * 256 + threadIdx.x;
  size_t bi = idx / n;             // = b*n + i
  int j = (int)(idx % n);
  size_t b = bi / n;
  float l = s2[bi] + d2[b * n + j];
  l = l >= 0.f ? l : 0.2f * l;
  l = adj[idx] ? l : NEGV;
  attnOut[idx] = __expf(l - Mrow[bi]) / Srow[bi];
}

// ---------------------------------------------------------------------------
extern "C" void kernel_launch(void* const* d_in, const int* in_sizes, int n_in,
                              void* d_out, int out_size, void* d_ws, size_t ws_size,
                              hipStream_t stream) {
  const float* x     = (const float*)d_in[0];
  const int*   adj   = (const int*)d_in[1];
  const float* idex  = (const float*)d_in[2];
  const float* w1    = (const float*)d_in[3];
  const float* asrc1 = (const float*)d_in[4];
  const float* adst1 = (const float*)d_in[5];
  const float* b1    = (const float*)d_in[6];
  const float* w2    = (const float*)d_in[7];
  const float* asrc2 = (const float*)d_in[8];
  const float* adst2 = (const float*)d_in[9];
  const float* b2    = (const float*)d_in[10];

  // Derive shapes: in_sizes[1] = bs*n*n, in_sizes[2] = bs*n, in_sizes[0] = bs*n*f
  int n  = in_sizes[1] / in_sizes[2];
  int bs = in_sizes[2] / n;
  int f  = in_sizes[0] / (bs * n);          // 64
  int H1 = in_sizes[3] / (f * f);           // 4
  int fo = 64;
  int F2 = H1 * fo;                         // 256

  float* ws = (float*)d_ws;
  size_t o = 0;
  float* h1     = ws + o; o += (size_t)bs * n * f;        // normed input
  float* hp1    = ws + o; o += (size_t)bs * H1 * n * fo;  // layer1 projections
  float* s1v    = ws + o; o += (size_t)bs * H1 * n;
  float* d1v    = ws + o; o += (size_t)bs * H1 * n;
  float* h2in   = ws + o; o += (size_t)bs * n * F2;       // elu(concat heads)
  float* h2norm = ws + o; o += (size_t)bs * n * F2;
  float* hp2    = ws + o; o += (size_t)bs * n * fo;
  float* s2v    = ws + o; o += (size_t)bs * n;
  float* d2v    = ws + o; o += (size_t)bs * n;
  float* M2     = ws + o; o += (size_t)bs * n;
  float* S2     = ws + o; o += (size_t)bs * n;

  float* outPtr  = (float*)d_out;                         // (bs,n,64)
  float* attnOut = (float*)d_out + (size_t)bs * n * fo;   // (bs,n,n)

  // 1) instance norm of x
  masked_norm<<<bs * f, 256, 0, stream>>>(x, idex, h1, n, f);
  // 2) hp1[b,h] = h1[b] @ w1[h]
  wmma_gemm_n64<<<dim3(n / 128, bs * H1), 256, 0, stream>>>(
      h1, w1, hp1, nullptr, n, f, H1, 0, (size_t)f * 64);
  // 3) s1/d1 per-row dots
  attn_sd<<<(bs * H1 * n) / 256, 256, 0, stream>>>(
      hp1, asrc1, adst1, s1v, d1v, bs * H1 * n, n, H1);
  // 4) fused layer-1 attention -> elu -> transposed concat
  gat1_flash<<<bs * H1 * (n / 16), 32, 0, stream>>>(
      hp1, s1v, d1v, adj, b1, h2in, n, H1);
  // 5) instance norm of layer-2 input (256 features)
  masked_norm<<<bs * F2, 256, 0, stream>>>(h2in, idex, h2norm, n, F2);
  // 6) hp2[b] = h2norm[b] @ w2[0]
  wmma_gemm_n64<<<dim3(n / 128, bs), 256, 0, stream>>>(
      h2norm, w2, hp2, nullptr, n, F2, 1, 0, (size_t)F2 * 64);
  // 7) s2/d2
  attn_sd<<<(bs * n) / 256, 256, 0, stream>>>(
      hp2, asrc2, adst2, s2v, d2v, bs * n, n, 1);
  // 8) layer-2 softmax row stats
  attn2_stats<<<bs * n, 256, 0, stream>>>(s2v, d2v, adj, M2, S2, n);
  // 9) materialize attn2 into d_out (required output)
  attn2_write<<<(int)(((size_t)bs * n * n) / 256), 256, 0, stream>>>(
      s2v, d2v, adj, M2, S2, attnOut, n);
  // 10) out = attn2 @ hp2 + b2  (B batched per-b: bStrideB = n*64)
  wmma_gemm_n64<<<dim3(n / 128, bs), 256, 0, stream>>>(
      attnOut, hp2, outPtr, b2, n, n, 1, (size_t)n * 64, 0);
}